// MPIdentity_54039278518614
// MI455X (gfx1250) — compile-verified
//
#include <hip/hip_runtime.h>
#include <math.h>

// MI455X / gfx1250, wave32. f32 end-to-end (matches reference numerics).
// Pipeline:
//   K1: hsum[b,h]   = sum_n h[b,n,h]                         (reduction, HBM-bound)
//   K2: message[b,:]= tanh(hsum@Wm^T + N*bm);  gimsg[b,g] = message@W_ih[:, :H]^T + b_ih
//   K3: main fused GRU: gh = h@W_hh^T via V_WMMA_F32_16X16X4_F32 (the 12.9-GFLOP GEMM),
//       + per-particle jets·W_ih[:,H:] (8-wide VALU dot), gate math, blend with h.

#define B_  1024
#define N_  128
#define H_  128
#define FE_ 8
#define G3_ (3 * H_)         // 384
#define ROWS_PER_BLK 64      // 4 M-tiles of 16 rows; all rows share one jet b

typedef __attribute__((ext_vector_type(2))) float v2f;
typedef __attribute__((ext_vector_type(8))) float v8f;

// ---------------- Kernel 1: per-jet column sums of h ----------------
__global__ __launch_bounds__(128) void hsum_kernel(const float* __restrict__ h,
                                                   float* __restrict__ hsum) {
    const int b = blockIdx.x;
    const int t = threadIdx.x;                 // h-column
    const float* p = h + b * (N_ * H_) + t;    // coalesced across t
    float s = 0.0f;
#pragma unroll 8
    for (int n = 0; n < N_; ++n) s += p[n * H_];
    hsum[b * H_ + t] = s;
}

// ------- Kernel 2: message = tanh(hsum@Wm^T + N*bm), gimsg = message@W_ih[:, :H]^T + b_ih -------
// 8 jets per block (weights re-read only 128x instead of 1024x).
__global__ __launch_bounds__(256) void msg_kernel(const float* __restrict__ hsum,
                                                  const float* __restrict__ Wm,
                                                  const float* __restrict__ bm,
                                                  const float* __restrict__ W_ih,
                                                  const float* __restrict__ b_ih,
                                                  float* __restrict__ gimsg) {
    __shared__ float hs[8 * H_];
    __shared__ float msg[8 * H_];
    const int tid = threadIdx.x;
    const int b0  = blockIdx.x * 8;

    for (int i = tid; i < 8 * H_; i += 256) hs[i] = hsum[b0 * H_ + i];
    __syncthreads();

    for (int idx = tid; idx < 8 * H_; idx += 256) {
        const int bb = idx >> 7, kk = idx & (H_ - 1);
        float acc = (float)N_ * bm[kk];        // dist accumulates N copies of bm
        const float* wrow = Wm + kk * H_;
        const float* hrow = hs + bb * H_;
#pragma unroll 4
        for (int c = 0; c < H_; ++c) acc = fmaf(hrow[c], wrow[c], acc);
        msg[idx] = tanhf(acc);
    }
    __syncthreads();

    for (int idx = tid; idx < 8 * G3_; idx += 256) {
        const int bb = idx / G3_, g = idx - bb * G3_;
        float acc = b_ih[g];
        const float* wrow = W_ih + g * (H_ + FE_);   // first H columns
        const float* mrow = msg + bb * H_;
#pragma unroll 4
        for (int c = 0; c < H_; ++c) acc = fmaf(mrow[c], wrow[c], acc);
        gimsg[(b0 + bb) * G3_ + g] = acc;
    }
}

// ---------------- Kernel 3: fused GRU update (WMMA f32) ----------------
// Block: 256 threads = 8 waves, 64 rows (4 M-tiles), all same jet b.
// Wave w owns output columns k in [16w, 16w+16) for gates r (g=k), z (g=H+k), n (g=2H+k).
__global__ __launch_bounds__(256) void gru_main_kernel(const float* __restrict__ h,
                                                       const float* __restrict__ jets,
                                                       const float* __restrict__ W_ih,
                                                       const float* __restrict__ W_hh,
                                                       const float* __restrict__ b_hh,
                                                       const float* __restrict__ gimsg,
                                                       float* __restrict__ out) {
    __shared__ float lds_h[ROWS_PER_BLK * H_];   // 32 KB of the 320 KB WGP LDS

    const int tid  = threadIdx.x;
    const int row0 = blockIdx.x * ROWS_PER_BLK;  // global particle row m = b*N + n
    const int b    = row0 >> 7;                  // 64-row blocks never cross a jet

    // Stage the 64x128 h tile (32 KB of A data reused by all 8 waves, 24 N-tiles)
    {
        const float4* src = (const float4*)(h + row0 * H_);
        float4* dst = (float4*)lds_h;
        for (int i = tid; i < ROWS_PER_BLK * H_ / 4; i += 256) dst[i] = src[i];
    }
    __syncthreads();

    const int lane = tid & 31;
    const int wave = tid >> 5;        // 0..7
    const int col  = lane & 15;       // A row (M) and B/D column (N) per WMMA layout
    const int hi   = lane >> 4;       // selects K pair 0..1 vs 2..3 within a step
    const int k    = wave * 16 + col; // output hidden index 0..127

    const v8f zero8 = {0.f, 0.f, 0.f, 0.f, 0.f, 0.f, 0.f, 0.f};
    v8f acc[4][3];
#pragma unroll
    for (int mt = 0; mt < 4; ++mt)
#pragma unroll
        for (int g = 0; g < 3; ++g) acc[mt][g] = zero8;

    // B-fragment base rows in W_hh (B[k][g] = W_hh[g][k])
    const float* whR = W_hh + (k         ) * H_;
    const float* whZ = W_hh + (k +     H_) * H_;
    const float* whN = W_hh + (k + 2 * H_) * H_;

    for (int kk = 0; kk < H_ / 4; ++kk) {        // 32 K-steps of K=4
        const int kb = kk * 4 + hi * 2;          // K layout: lanes16-31 carry K+2,K+3
        const v2f bR = *(const v2f*)(whR + kb);
        const v2f bZ = *(const v2f*)(whZ + kb);
        const v2f bN = *(const v2f*)(whN + kb);
#pragma unroll
        for (int mt = 0; mt < 4; ++mt) {
            const v2f a = *(const v2f*)(&lds_h[(mt * 16 + col) * H_ + kb]);
            acc[mt][0] = __builtin_amdgcn_wmma_f32_16x16x4_f32(
                false, a, false, bR, (short)0, acc[mt][0], false, false);
            acc[mt][1] = __builtin_amdgcn_wmma_f32_16x16x4_f32(
                false, a, false, bZ, (short)0, acc[mt][1], false, false);
            acc[mt][2] = __builtin_amdgcn_wmma_f32_16x16x4_f32(
                false, a, false, bN, (short)0, acc[mt][2], false, false);
        }
    }

    // Per-lane uniform epilogue operands (same for all rows of the block)
    const float gm_r = gimsg[b * G3_ + k];
    const float gm_z = gimsg[b * G3_ + k + H_];
    const float gm_n = gimsg[b * G3_ + k + 2 * H_];
    const float bh_r = b_hh[k];
    const float bh_z = b_hh[k + H_];
    const float bh_n = b_hh[k + 2 * H_];

    float wjr[FE_], wjz[FE_], wjn[FE_];          // W_ih jet columns for this lane's k
#pragma unroll
    for (int f = 0; f < FE_; ++f) {
        wjr[f] = W_ih[(k         ) * (H_ + FE_) + H_ + f];
        wjz[f] = W_ih[(k +     H_) * (H_ + FE_) + H_ + f];
        wjn[f] = W_ih[(k + 2 * H_) * (H_ + FE_) + H_ + f];
    }

#pragma unroll
    for (int mt = 0; mt < 4; ++mt) {
#pragma unroll
        for (int r = 0; r < 8; ++r) {
            // D layout: VGPR r holds (M=r, N=col) for lanes 0-15, (M=r+8) for 16-31
            const int mloc = mt * 16 + r + hi * 8;
            const int m    = row0 + mloc;

            float jr = gm_r, jz = gm_z, jn = gm_n;    // i_* gate pre-activations
            const float* jrow = jets + m * FE_;
#pragma unroll
            for (int f = 0; f < FE_; ++f) {
                const float jv = jrow[f];
                jr = fmaf(jv, wjr[f], jr);
                jz = fmaf(jv, wjz[f], jz);
                jn = fmaf(jv, wjn[f], jn);
            }

            const float hr = acc[mt][0][r] + bh_r;    // h_* gate pre-activations
            const float hz = acc[mt][1][r] + bh_z;
            const float hn = acc[mt][2][r] + bh_n;

            const float rg = 1.0f / (1.0f + expf(-(jr + hr)));
            const float zg = 1.0f / (1.0f + expf(-(jz + hz)));
            const float ng = tanhf(fmaf(rg, hn, jn));

            const float hval = lds_h[mloc * H_ + k];
            out[m * H_ + k] = (1.0f - zg) * ng + zg * hval;
        }
    }
}

extern "C" void kernel_launch(void* const* d_in, const int* in_sizes, int n_in,
                              void* d_out, int out_size, void* d_ws, size_t ws_size,
                              hipStream_t stream) {
    const float* h    = (const float*)d_in[0];
    const float* jets = (const float*)d_in[1];
    // d_in[2] = mask (unused by reference forward)
    const float* Wm   = (const float*)d_in[3];
    const float* bm   = (const float*)d_in[4];
    const float* W_ih = (const float*)d_in[5];
    const float* W_hh = (const float*)d_in[6];
    const float* b_ih = (const float*)d_in[7];
    const float* b_hh = (const float*)d_in[8];
    float* out = (float*)d_out;

    // Workspace: hsum [B*H] then gimsg [B*3H]  (2 MB total)
    float* hsum  = (float*)d_ws;
    float* gimsg = hsum + B_ * H_;

    hsum_kernel<<<B_, 128, 0, stream>>>(h, hsum);
    msg_kernel<<<B_ / 8, 256, 0, stream>>>(hsum, Wm, bm, W_ih, b_ih, gimsg);
    gru_main_kernel<<<(B_ * N_) / ROWS_PER_BLK, 256, 0, stream>>>(
        h, jets, W_ih, W_hh, b_hh, gimsg, out);
}